// GrapheneModel_32530082300274
// MI455X (gfx1250) — compile-verified
//
#include <hip/hip_runtime.h>

// CDNA5 / gfx1250 graphene band-structure kernel, v2.
//
// Per k-point: d1 = k.delta1, d2 = k.delta2 via V_WMMA_F32_16X16X4_F32.
// All nine reference phases reduce algebraically to (d1, d2):
//   d0 = -(d1+d2); nnn cosines: cos(d1-d2), cos(2d1+d2), cos(d1+2d2) (x2 by +/-)
//   f_re = -t (cos(d1+d2) + cos d1 + cos d2)
//   f_im = -t (sin d1 + sin d2 - sin(d1+d2))
//   e    = -2 t' (cos(d1-d2) + cos(2d1+d2) + cos(d1+2d2))
//   out  = (e -/+ sqrt(m^2 + f_re^2 + f_im^2)) * EV_TO_J
//
// Two WMMAs per wave cover 32 distinct points: WMMA#0 carries the delta
// vectors in A rows {0,1} (D rows 0,1 -> lanes 0-15), WMMA#1 in rows {8,9}
// (D rows 8,9 -> lanes 16-31). Each lane then owns one point: 4 trig + 1 raw
// v_sqrt per point, and stores its (e-r, e+r) pair as one b64 (wave = 256 B
// contiguous). Bandwidth-bound: 16 B/point, 64 MB total ~ 2.9 us @ 23.3 TB/s.

typedef float v2f __attribute__((ext_vector_type(2)));
typedef float v8f __attribute__((ext_vector_type(8)));

__global__ __launch_bounds__(256) void
graphene_bands_wmma(const float* __restrict__ kpts,
                    float* __restrict__ out,
                    int npts) {
    const int lane   = threadIdx.x & 31;
    const int wave   = threadIdx.x >> 5;
    const int lane15 = lane & 15;
    const bool lo_half = (lane < 16);

    // this wave covers points [base, base+32); each lane owns point base+lane
    const int base = (blockIdx.x * (blockDim.x >> 5) + wave) * 32;
    const int p    = base + lane;

    // delta1 = ( a_lat/2, a_cc/2 ), delta2 = ( -a_lat/2, a_cc/2 )  [meters]
    const float D1X = 1.23e-10f;          // a_lat / 2
    const float DY  = 7.10140635e-11f;    // a_cc  / 2

    // ---- A operands (16x4 f32, 2 VGPRs: VGPR0=K0/K2, VGPR1=K1/K3) ----
    // A0: rows {0,1} = delta1, delta2 -> D rows 0,1 (lanes 0-15)
    // A1: rows {8,9} = delta1, delta2 -> D rows 8,9 (lanes 16-31)
    v2f a0 = {0.0f, 0.0f};
    v2f a1 = {0.0f, 0.0f};
    if (lo_half) {
        if (lane == 0) { a0.x =  D1X; a0.y = DY; }
        if (lane == 1) { a0.x = -D1X; a0.y = DY; }
        if (lane == 8) { a1.x =  D1X; a1.y = DY; }
        if (lane == 9) { a1.x = -D1X; a1.y = DY; }
    }

    // ---- B operands (4x16 f32): column n = (kx_n, ky_n, 0, 0) ----
    // lanes 0-15 carry K0/K1 rows; lanes 16-31 (K2/K3) stay zero.
    v2f b0 = {0.0f, 0.0f};
    v2f b1 = {0.0f, 0.0f};
    if (lo_half) {
        const int q0 = base + lane15;        // group 0 point
        const int q1 = base + 16 + lane15;   // group 1 point
        if (q0 < npts) b0 = ((const v2f*)kpts)[q0];
        if (q1 < npts) b1 = ((const v2f*)kpts)[q1];
    }

    // ---- two back-to-back WMMAs (EXEC all-1s here) ----
    v8f c = {};
    v8f d0 = __builtin_amdgcn_wmma_f32_16x16x4_f32(
        false, a0, false, b0, (short)0, c, false, false);
    v8f d1g = __builtin_amdgcn_wmma_f32_16x16x4_f32(
        false, a1, false, b1, (short)0, c, false, false);

    // lanes 0-15: own point's phases in d0 rows 0,1; lanes 16-31: d1g rows 8,9
    const float d1 = lo_half ? d0[0] : d1g[0];
    const float d2 = lo_half ? d0[1] : d1g[1];

    // ---- 4 transcendentals per point; rest is FMA algebra ----
    const float s1 = __sinf(d1), c1 = __cosf(d1);
    const float s2 = __sinf(d2), c2 = __cosf(d2);

    const float C12 = c1 * c2 - s1 * s2;   // cos(d1+d2)
    const float S12 = s1 * c2 + c1 * s2;   // sin(d1+d2)

    const float T_EV  = 2.8f;
    const float TP_EV = 0.1f;
    const float M2    = 0.0025f;           // (0.05 eV)^2
    const float EV2J  = 1.602176634e-19f;

    const float f_re = -T_EV * (C12 + c1 + c2);
    const float f_im = -T_EV * (s1 + s2 - S12);

    const float cos2u = c1 * c2 + s1 * s2;       // cos(d1-d2)
    const float cosA  = c1 * C12 - s1 * S12;     // cos(2*d1 +   d2)
    const float cosB  = C12 * c2 - S12 * s2;     // cos(  d1 + 2*d2)
    const float e     = -2.0f * TP_EV * (cos2u + cosA + cosB);

    // raw v_sqrt_f32: arg in [2.5e-3, ~71], no IEEE fixup sequence needed
    const float r = __builtin_amdgcn_sqrtf(fmaf(f_re, f_re, fmaf(f_im, f_im, M2)));

    v2f res;
    res.x = (e - r) * EV2J;
    res.y = (e + r) * EV2J;

    if (p < npts) {
        ((v2f*)out)[p] = res;   // wave stores 256 contiguous bytes
    }
}

extern "C" void kernel_launch(void* const* d_in, const int* in_sizes, int n_in,
                              void* d_out, int out_size, void* d_ws, size_t ws_size,
                              hipStream_t stream) {
    (void)n_in; (void)d_ws; (void)ws_size; (void)out_size;
    const float* k = (const float*)d_in[0];
    float* out = (float*)d_out;
    const int npts = in_sizes[0] / 2;             // k is [N,2] interleaved

    const int threads = 256;                      // 8 waves -> 256 points / block
    const int pts_per_block = (threads >> 5) * 32;
    const int blocks = (npts + pts_per_block - 1) / pts_per_block;

    graphene_bands_wmma<<<blocks, threads, 0, stream>>>(k, out, npts);
}